// CustomGATLayer_90348932038855
// MI455X (gfx1250) — compile-verified
//
#include <hip/hip_runtime.h>
#include <hip/hip_bf16.h>

// GAT layer on MI455X (gfx1250), wave32 + WMMA f32 16x16x4.
//
// scores are rank-1 (e_i + e_j) so the only GEMMs are X@W and attn@Wh, both
// mapped to v_wmma_f32_16x16x4_f32. Softmax stability uses a precomputed
// per-row UPPER BOUND m_row = leaky(e_i + max_j e_j) (leaky is monotone), so
// the attention inner loop needs NO online max, NO accumulator rescale and NO
// cross-lane shuffles: it is pure {load, exp(TRANS), wmma(XDL)} which the ISA
// says co-execute. HBM traffic is dominated by the 32MB adjacency stream
// (~1.4us floor @ 23.3 TB/s); Wh/e vectors stay resident in the 192MB L2.

typedef float v2f __attribute__((ext_vector_type(2)));
typedef float v8f __attribute__((ext_vector_type(8)));

#define B_    8
#define N_    1024
#define D_    256
#define H_    8
#define HD_   32
#define NEGINF_ (-1000000000.0f)
#define SLOPE_  0.2f

// ---------------------------------------------------------------------------
// Kernel 1: Wh[b,h,n,j] = X[b,n,:] @ W[h,:,j]   (per-(b,h): 1024x256 @ 256x32)
// One wave per 16-row tile. A-layout: lane l holds row n0+l, VGPR pair holds
// K = k0 + 2*half + {0,1}. Two 16x16 C tiles cover hd=32 columns.
// ---------------------------------------------------------------------------
__global__ __launch_bounds__(128) void gat_wh_kernel(
    const float* __restrict__ X, const float* __restrict__ W,
    float* __restrict__ Wh) {
  const int lane = threadIdx.x & 31;
  const int wave = threadIdx.x >> 5;
  const int t  = blockIdx.x * 4 + wave;   // 0..4095 tiles = B*H*(N/16)
  const int nt = t & 63;
  const int h  = (t >> 6) & 7;
  const int b  = t >> 9;
  const int n0 = nt * 16;
  const int half = lane >> 4;
  const int l    = lane & 15;

  const float* xrow = X + (size_t)(b * N_ + n0 + l) * D_;
  const float* wmat = W + (size_t)h * D_ * HD_;

  v8f c0 = {}; v8f c1 = {};
  #pragma unroll 4
  for (int k0 = 0; k0 < D_; k0 += 4) {
    const int ka = k0 + 2 * half;                 // K pair base for this half-wave
    v2f a = *(const v2f*)(xrow + ka);             // A: rows x K (global_load_b64)
    v2f b0, b1;                                   // B: K x cols (rows of W)
    b0.x = wmat[(size_t)ka * HD_ + l];
    b0.y = wmat[(size_t)(ka + 1) * HD_ + l];
    b1.x = wmat[(size_t)ka * HD_ + l + 16];
    b1.y = wmat[(size_t)(ka + 1) * HD_ + l + 16];
    c0 = __builtin_amdgcn_wmma_f32_16x16x4_f32(false, a, false, b0, (short)0, c0, false, false);
    c1 = __builtin_amdgcn_wmma_f32_16x16x4_f32(false, a, false, b1, (short)0, c1, false, false);
  }

  // C layout: VGPR r -> row r (lanes 0-15) / row r+8 (lanes 16-31), col = lane
  float* whb = Wh + (size_t)(b * H_ + h) * N_ * HD_;
  #pragma unroll
  for (int r = 0; r < 8; ++r) {
    const int row = n0 + r + 8 * half;
    whb[(size_t)row * HD_ + l]      = c0[r];
    whb[(size_t)row * HD_ + l + 16] = c1[r];
  }
}

// ---------------------------------------------------------------------------
// Kernel 2: e1[row] = Wh[row,:] . a1[h],  e2[row] = Wh[row,:] . a2[h]
// ---------------------------------------------------------------------------
__global__ void gat_e_kernel(const float* __restrict__ Wh,
                             const float* __restrict__ avec,
                             float* __restrict__ e1, float* __restrict__ e2) {
  const int idx = blockIdx.x * blockDim.x + threadIdx.x;
  if (idx >= B_ * H_ * N_) return;
  const int h = (idx / N_) & (H_ - 1);
  const float* wrow = Wh + (size_t)idx * HD_;
  const float* a1 = avec + (size_t)h * 2 * HD_;
  const float* a2 = a1 + HD_;
  float s1 = 0.f, s2 = 0.f;
  #pragma unroll
  for (int j = 0; j < HD_; ++j) {
    const float w = wrow[j];
    s1 += w * a1[j];
    s2 += w * a2[j];
  }
  e1[idx] = s1;
  e2[idx] = s2;
}

// ---------------------------------------------------------------------------
// Kernel 2b: emax[b,h] = max_n e2[b,h,n].  One wave per (b,h).
// ---------------------------------------------------------------------------
__global__ __launch_bounds__(32) void gat_emax_kernel(
    const float* __restrict__ e2, float* __restrict__ emax) {
  const int bh   = blockIdx.x;          // 0..63
  const int lane = threadIdx.x;         // 0..31
  const float* p = e2 + (size_t)bh * N_;
  float mx = -3.0e38f;
  #pragma unroll
  for (int k = 0; k < N_ / 32; ++k) mx = fmaxf(mx, p[lane + 32 * k]);
  #pragma unroll
  for (int s = 16; s > 0; s >>= 1) mx = fmaxf(mx, __shfl_xor(mx, s, 32));
  if (lane == 0) emax[bh] = mx;
}

// ---------------------------------------------------------------------------
// Kernel 3: fused masked-leaky-softmax + attn @ Wh.
// One wave per 16-row output tile; stream j in blocks of 16.
// m_row is a precomputed upper bound -> inner loop has zero cross-lane ops:
// per block it is 24 loads, 8 exp (TRANS), 8 v_wmma (XDL).
// P built directly in the WMMA A-operand layout:
//   lane l (both halves) owns row i0+l; half selects K parity {2h,2h+1} mod 4.
// ---------------------------------------------------------------------------
__global__ __launch_bounds__(256) void gat_attn_kernel(
    const int* __restrict__ adj, const float* __restrict__ Wh,
    const float* __restrict__ e1, const float* __restrict__ e2,
    const float* __restrict__ emax, const float* __restrict__ bias,
    float* __restrict__ out) {
  const int lane = threadIdx.x & 31;
  const int wave = threadIdx.x >> 5;
  const int t  = blockIdx.x * 8 + wave;   // 0..4095 tiles
  const int it = t & 63;
  const int h  = (t >> 6) & 7;
  const int b  = t >> 9;
  const int i0 = it * 16;
  const int half = lane >> 4;
  const int l    = lane & 15;

  const size_t bh = (size_t)(b * H_ + h);
  const float  ei   = e1[bh * N_ + i0 + l];
  const float* e2p  = e2 + bh * N_;
  const float* whb  = Wh + bh * N_ * HD_;
  const int*   arow = adj + ((size_t)b * N_ + (i0 + l)) * N_;

  // fixed per-row softmax bound: leaky is monotone, so
  // leaky(ei + ej) <= leaky(ei + max ej); masked entries are NEGINF_.
  float mrow = ei + emax[bh];
  mrow = fmaxf(mrow, SLOPE_ * mrow);
  mrow = fmaxf(mrow, NEGINF_);

  float lsum = 0.0f;        // lane-local partial row sum
  v8f acc0 = {}; v8f acc1 = {};

  for (int j0 = 0; j0 < N_; j0 += 16) {
    if (j0 + 16 < N_) __builtin_prefetch(arow + j0 + 16, 0, 1);

    float p[8];
    #pragma unroll
    for (int k = 0; k < 4; ++k) {
      const int jj = j0 + 4 * k + 2 * half;
      v2f  ej = *(const v2f*)(e2p + jj);
      int2 ad = *(const int2*)(arow + jj);
      float s0 = ei + ej.x; s0 = fmaxf(s0, SLOPE_ * s0); s0 = ad.x ? s0 : NEGINF_;
      float s1 = ei + ej.y; s1 = fmaxf(s1, SLOPE_ * s1); s1 = ad.y ? s1 : NEGINF_;
      const float p0 = __expf(s0 - mrow);
      const float p1 = __expf(s1 - mrow);
      p[2 * k]     = p0;
      p[2 * k + 1] = p1;
      lsum += p0 + p1;
    }

    // acc += P(16x16) @ Wh(16x32): 4 K-steps x 2 N-tiles of wmma f32 16x16x4
    #pragma unroll
    for (int k = 0; k < 4; ++k) {
      const int jr = j0 + 4 * k + 2 * half;
      v2f pa; pa.x = p[2 * k]; pa.y = p[2 * k + 1];
      v2f b0, b1;
      b0.x = whb[(size_t)jr * HD_ + l];
      b0.y = whb[(size_t)(jr + 1) * HD_ + l];
      b1.x = whb[(size_t)jr * HD_ + l + 16];
      b1.y = whb[(size_t)(jr + 1) * HD_ + l + 16];
      acc0 = __builtin_amdgcn_wmma_f32_16x16x4_f32(false, pa, false, b0, (short)0, acc0, false, false);
      acc1 = __builtin_amdgcn_wmma_f32_16x16x4_f32(false, pa, false, b1, (short)0, acc1, false, false);
    }
  }

  // epilogue: reduce row sums across halves (lanes l <-> l+16 share row l),
  // broadcast per C-tile row (8*half + r), normalize, bias, relu, store.
  const float lr = lsum + __shfl_xor(lsum, 16, 32);
  #pragma unroll
  for (int r = 0; r < 8; ++r) {
    const float ls  = __shfl(lr, 8 * half + r, 32);
    const float inv = 1.0f / fmaxf(ls, 1.0e-30f);
    const int row = i0 + r + 8 * half;
    float* orow = out + ((size_t)b * N_ + row) * (H_ * HD_) + h * HD_;
    orow[l]      = fmaxf(acc0[r] * inv + bias[h * HD_ + l],      0.0f);
    orow[l + 16] = fmaxf(acc1[r] * inv + bias[h * HD_ + l + 16], 0.0f);
  }
}

// ---------------------------------------------------------------------------
extern "C" void kernel_launch(void* const* d_in, const int* in_sizes, int n_in,
                              void* d_out, int out_size, void* d_ws, size_t ws_size,
                              hipStream_t stream) {
  (void)in_sizes; (void)n_in; (void)out_size; (void)ws_size;
  const float* X    = (const float*)d_in[0];   // (B,N,D) f32
  const int*   adj  = (const int*)d_in[1];     // (B,N,N) i32
  const float* W    = (const float*)d_in[2];   // (H,D,hd) f32
  const float* avec = (const float*)d_in[3];   // (H,2*hd) f32
  const float* bias = (const float*)d_in[4];   // (H*hd,) f32
  float* out = (float*)d_out;                  // (B,N,H*hd) f32

  // workspace: Wh (8MB) + e1/e2 (256KB each) + emax (64 floats)
  float* Wh   = (float*)d_ws;
  float* e1   = Wh + (size_t)B_ * H_ * N_ * HD_;
  float* e2   = e1 + (size_t)B_ * H_ * N_;
  float* emax = e2 + (size_t)B_ * H_ * N_;

  gat_wh_kernel<<<1024, 128, 0, stream>>>(X, W, Wh);                       // 4096 waves
  gat_e_kernel<<<(B_ * H_ * N_ + 255) / 256, 256, 0, stream>>>(Wh, avec, e1, e2);
  gat_emax_kernel<<<B_ * H_, 32, 0, stream>>>(e2, emax);
  gat_attn_kernel<<<512, 256, 0, stream>>>(adj, Wh, e1, e2, emax, bias, out); // 4096 waves
}